// Mandala2d_67628555043114
// MI455X (gfx1250) — compile-verified
//
#include <hip/hip_runtime.h>
#include <math.h>

#define H 224
#define W 224
#define HW (H * W)        // 50176
#define BATCH 16
#define CHANNELS 64
#define NRINGS 79         // md = round(sqrt(2)*112) = 158; ceil(158/2) = 79
#define MAXL 2048         // largest ring ~1400 pixels; pad to pow2 for bitonic

// ---------------------------------------------------------------------------
// Exact angle comparator. angle = atan2(j-111.5, i-111.5) + 180.
// Scale coords by 2: X = 2i-223, Y = 2j-223 (odd ints, never 0 -> no axis
// ambiguity). Order: Y<0 half (angle in (0,180)) before Y>0 half; within a
// half-plane, theta_a < theta_b  <=>  cross(a,b) = Xa*Yb - Ya*Xb > 0.
// Exact-collinear ties (the only ties double atan2 produces) break by index.
// 0xFFFFFFFF acts as +inf padding.
// ---------------------------------------------------------------------------
__device__ __forceinline__ bool angLess(unsigned a, unsigned b) {
    if (b == 0xFFFFFFFFu) return a != 0xFFFFFFFFu;
    if (a == 0xFFFFFFFFu) return false;
    int ia = (int)(a / W), ja = (int)(a % W);
    int ib = (int)(b / W), jb = (int)(b % W);
    int Xa = 2 * ia - (W - 1), Ya = 2 * ja - (W - 1);
    int Xb = 2 * ib - (W - 1), Yb = 2 * jb - (W - 1);
    int ha = (Ya > 0) ? 1 : 0;
    int hb = (Yb > 0) ? 1 : 0;
    if (ha != hb) return ha < hb;
    long long cr = (long long)Xa * (long long)Yb - (long long)Ya * (long long)Xb;
    if (cr != 0) return cr > 0;
    return a < b;
}

// ---------------------------------------------------------------------------
// Kernel A: build angle-sorted pixel list per ring. One block per ring.
// ---------------------------------------------------------------------------
__global__ __launch_bounds__(1024) void ring_build_kernel(unsigned* __restrict__ ringPix,
                                                          int* __restrict__ ringLen) {
    __shared__ unsigned s_ids[MAXL];
    __shared__ int s_cnt;
    const int tid  = threadIdx.x;
    const int ring = blockIdx.x;

    if (tid == 0) s_cnt = 0;
    for (int p = tid; p < MAXL; p += 1024) s_ids[p] = 0xFFFFFFFFu;
    __syncthreads();

    for (int p = tid; p < HW; p += 1024) {
        int i = p / W, j = p % W;
        float di = (float)i - 111.5f;
        float dj = (float)j - 111.5f;
        float d  = sqrtf(di * di + dj * dj);
        int k = (int)(d * 0.5f);          // ring = floor(d/2); boundary-safe (d^2 = int + 0.5)
        if (k == ring) {
            int slot = atomicAdd(&s_cnt, 1);   // LDS atomic
            if (slot < MAXL) s_ids[slot] = (unsigned)p;
        }
    }
    __syncthreads();

    // Bitonic sort of 2048 slots with 1024 threads (one compare-exchange each).
    for (unsigned size = 2; size <= MAXL; size <<= 1) {
        for (unsigned stride = size >> 1; stride > 0; stride >>= 1) {
            unsigned t   = (unsigned)tid;
            unsigned pos = 2u * t - (t & (stride - 1u));
            unsigned va  = s_ids[pos];
            unsigned vb  = s_ids[pos + stride];
            bool asc = ((pos & size) == 0u);
            if (angLess(vb, va) == asc) {
                s_ids[pos]          = vb;
                s_ids[pos + stride] = va;
            }
            __syncthreads();
        }
    }

    int L = s_cnt;
    if (L > MAXL) L = MAXL;
    if (tid == 0) ringLen[ring] = L;
    for (int p = tid; p < MAXL; p += 1024) ringPix[ring * MAXL + p] = s_ids[p];
}

// ---------------------------------------------------------------------------
// Kernel B: per (ring, batch) first-occurrence argmax over channel 0.
// ---------------------------------------------------------------------------
__global__ __launch_bounds__(256) void ring_argmax_kernel(const float* __restrict__ x,
                                                          const unsigned* __restrict__ ringPix,
                                                          const int* __restrict__ ringLen,
                                                          int* __restrict__ shiftTab) {
    const int ring = blockIdx.x;
    const int b    = blockIdx.y;
    const int tid  = threadIdx.x;
    const int L    = ringLen[ring];
    const unsigned* pix = ringPix + ring * MAXL;
    const float* plane0 = x + (size_t)b * CHANNELS * HW;   // channel 0 plane

    float best = -INFINITY;
    int   bi   = 0x7fffffff;
    for (int p = tid; p < L; p += 256) {
        float v = plane0[pix[p]];
        if (v > best) { best = v; bi = p; }   // strict '>' keeps first occurrence per thread
    }

    __shared__ float sv[256];
    __shared__ int   si[256];
    sv[tid] = best; si[tid] = bi;
    __syncthreads();
    for (int s = 128; s > 0; s >>= 1) {
        if (tid < s) {
            float v2 = sv[tid + s];
            int   i2 = si[tid + s];
            if (v2 > sv[tid] || (v2 == sv[tid] && i2 < si[tid])) { sv[tid] = v2; si[tid] = i2; }
        }
        __syncthreads();
    }
    if (tid == 0) shiftTab[b * NRINGS + ring] = si[0];
}

// ---------------------------------------------------------------------------
// Async global->LDS gather (CDNA5 path): per-lane scattered global addresses,
// data lands in LDS, tracked by ASYNCcnt (async loads drain in issue order).
// ---------------------------------------------------------------------------
__device__ __forceinline__ void asyncLoadToLds(unsigned lds_addr, const float* gaddr) {
    asm volatile("global_load_async_to_lds_b32 %0, %1, off"
                 :: "v"(lds_addr), "v"(gaddr)
                 : "memory");
}

// ---------------------------------------------------------------------------
// Kernel C: per (ring, batch) cyclic roll of all 64 channels.
//   out[b,c,pix[p]] = x[b,c,pix[(p+shift) % L]]
// Pipeline: 4 LDS buffers, 3 async gather tiles in flight. Every wave issues
// EVERY tile (address clamped for pos>=L) so per-wave ASYNCcnt accounting is
// uniform; the wait immediate equals the number of loads issued after tile t
// (min(3, total-1-t)), which with in-order load drain guarantees tile t's LDS
// data is resident — including the tail tiles.
// ---------------------------------------------------------------------------
__global__ __launch_bounds__(256) void ring_roll_kernel(const float* __restrict__ x,
                                                        float* __restrict__ out,
                                                        const unsigned* __restrict__ ringPix,
                                                        const int* __restrict__ ringLen,
                                                        const int* __restrict__ shiftTab) {
    const int ring  = blockIdx.x;
    const int b     = blockIdx.y;
    const int tid   = threadIdx.x;
    const int L     = ringLen[ring];
    const int shift = shiftTab[b * NRINGS + ring];
    const unsigned* pix = ringPix + ring * MAXL;

    __shared__ int   s_src[MAXL];
    __shared__ int   s_dst[MAXL];
    __shared__ float sbuf[4][256];

    // Stage rolled src / dst pixel tables once per block (reused for 64 channels).
    for (int p = tid; p < L; p += 256) {
        s_dst[p] = (int)pix[p];
        int q = p + shift;
        if (q >= L) q -= L;
        s_src[p] = (int)pix[q];
    }
    __syncthreads();

    const size_t baseBC = (size_t)b * CHANNELS * HW;
    const int nt    = (L + 255) >> 8;     // tiles of 256 per channel
    const int total = nt * CHANNELS;

    // LDS byte addresses of this lane's 4 staging slots (low 32 bits of the
    // generic shared-aperture address == workgroup-relative LDS offset).
    const unsigned ldsBase = (unsigned)(unsigned long long)&sbuf[0][tid];

    // Unconditional issue of tile q's gather into buffer q&3 (pos clamped).
    auto issueTile = [&](int q) {
        int c    = q / nt;
        int tile = q - c * nt;
        int pos  = (tile << 8) + tid;
        int posc = (pos < L) ? pos : (L - 1);
        asyncLoadToLds(ldsBase + (unsigned)((q & 3) << 10),
                       x + baseBC + (size_t)c * HW + (size_t)s_src[posc]);
    };

    // Prologue: fill the pipeline (up to 3 tiles in flight).
    for (int q = 0; q < 3 && q < total; ++q) issueTile(q);

    for (int t = 0; t < total; ++t) {
        if (t + 3 < total) issueTile(t + 3);

        // Exactly min(3, total-1-t) loads were issued after tile t by every wave.
        const int rem = total - 1 - t;
        if (rem >= 3)      asm volatile("s_wait_asynccnt 0x3" ::: "memory");
        else if (rem == 2) asm volatile("s_wait_asynccnt 0x2" ::: "memory");
        else if (rem == 1) asm volatile("s_wait_asynccnt 0x1" ::: "memory");
        else               asm volatile("s_wait_asynccnt 0x0" ::: "memory");

        int c    = t / nt;
        int tile = t - c * nt;
        int pos  = (tile << 8) + tid;
        if (pos < L) {
            float v = sbuf[t & 3][tid];                       // ds_load_b32
            out[baseBC + (size_t)c * HW + (size_t)s_dst[pos]] = v;
        }
    }
}

// ---------------------------------------------------------------------------
// Host entry. Workspace layout: ringPix[79*2048] u32 | ringLen[79] i32 |
// shiftTab[16*79] i32  (< 700 KB total).
// ---------------------------------------------------------------------------
extern "C" void kernel_launch(void* const* d_in, const int* in_sizes, int n_in,
                              void* d_out, int out_size, void* d_ws, size_t ws_size,
                              hipStream_t stream) {
    (void)in_sizes; (void)n_in; (void)out_size; (void)ws_size;
    const float* x = (const float*)d_in[0];
    float* out     = (float*)d_out;

    unsigned* ringPix  = (unsigned*)d_ws;
    int*      ringLen  = (int*)(ringPix + NRINGS * MAXL);
    int*      shiftTab = ringLen + NRINGS;

    ring_build_kernel<<<NRINGS, 1024, 0, stream>>>(ringPix, ringLen);
    ring_argmax_kernel<<<dim3(NRINGS, BATCH), 256, 0, stream>>>(x, ringPix, ringLen, shiftTab);
    ring_roll_kernel<<<dim3(NRINGS, BATCH), 256, 0, stream>>>(x, out, ringPix, ringLen, shiftTab);
}